// GNN_nl_43679817400582
// MI455X (gfx1250) — compile-verified
//
#include <hip/hip_runtime.h>
#include <math.h>

// ---------------------------------------------------------------------------
// GNN forward for MI455X (gfx1250), fp32 precision via V_WMMA_F32_16X16X4_F32.
// Dominant GEMM: per-batch D(676 x fin) @ W1^T(fin x 192), fin in {128..320}.
// BN batch-stats force a 2-pass structure; H is recomputed (not materialized).
// ---------------------------------------------------------------------------

typedef __attribute__((ext_vector_type(2))) float v2f;
typedef __attribute__((ext_vector_type(8))) float v8f;

#define B_      512
#define N_      26
#define F0_     128
#define HALF_   48
#define XMAX    320          // max feature width (layer-append layout stride)
#define CCH     192          // 2*NF channels in wcompute
#define ROWS    676          // N*N (i,j) pairs
#define RT      43           // ceil(676/16) row tiles
#define EPS_    1e-5f
#define SLOPE   0.01f
#define NTHREADS 384         // 12 wave32 = 12 channel tiles of 16 -> 192 ch

// ---------------------------------------------------------------------------
__global__ void k_copy(const float* __restrict__ x, float* __restrict__ X) {
    int idx = blockIdx.x * blockDim.x + threadIdx.x;
    if (idx < B_ * N_ * F0_) {
        int f = idx % F0_;
        int bn = idx / F0_;
        X[bn * XMAX + f] = x[idx];
    }
}

__global__ void k_zero(float* __restrict__ p, int n) {
    int i = blockIdx.x * blockDim.x + threadIdx.x;
    if (i < n) p[i] = 0.f;
}

// ---------------------------------------------------------------------------
// Pass 1: compute H = |x_i - x_j| @ W1^T tile-by-tile with f32 WMMA, reduce
// per-channel sum / sumsq for BN batch statistics. Block = (b, row-tile),
// 12 waves each own a 16-channel column tile.
// ---------------------------------------------------------------------------
__global__ void __launch_bounds__(NTHREADS)
k_wstats(const float* __restrict__ X, const float* __restrict__ w1,
         const float* __restrict__ b1, int fin,
         float* __restrict__ wsum, float* __restrict__ wsumsq)
{
    __shared__ float Xs[N_ * (XMAX + 4)];   // padded stride vs LDS banks
    __shared__ float st[2 * CCH];

    const int tid = threadIdx.x;
    const int b   = blockIdx.x / RT;
    const int rt  = blockIdx.x % RT;
    const int xstr = fin + 4;

    const float* xb = X + (size_t)b * N_ * XMAX;
    for (int idx = tid; idx < N_ * fin; idx += NTHREADS) {
        int n = idx / fin, f = idx - n * fin;
        Xs[n * xstr + f] = xb[n * XMAX + f];
    }
    for (int i = tid; i < 2 * CCH; i += NTHREADS) st[i] = 0.f;
    __syncthreads();

    const int wave  = tid >> 5;        // channel tile 0..11
    const int lane  = tid & 31;
    const int l16   = lane & 15;
    const int khalf = lane >> 4;       // K pair select (0: K0/K1, 1: K2/K3)

    // A fragment row (same 16 rows for both lane halves)
    int grow = rt * 16 + l16;
    int gr   = (grow < ROWS) ? grow : 0;      // padded rows -> i==j -> d==0
    int ii   = gr / N_, jj = gr - ii * N_;
    const float* xi = Xs + ii * xstr;
    const float* xj = Xs + jj * xstr;

    const int ch = wave * 16 + l16;           // B fragment column = channel
    const float* wrow = w1 + (size_t)ch * fin;

    v8f c = {0.f, 0.f, 0.f, 0.f, 0.f, 0.f, 0.f, 0.f};
    for (int k = 0; k < fin; k += 4) {
        int f0 = k + 2 * khalf;
        v2f a, bb;
        a.x  = fabsf(xi[f0]     - xj[f0]);
        a.y  = fabsf(xi[f0 + 1] - xj[f0 + 1]);
        bb.x = wrow[f0];
        bb.y = wrow[f0 + 1];
        c = __builtin_amdgcn_wmma_f32_16x16x4_f32(false, a, false, bb,
                                                  (short)0, c, false, false);
    }

    // per-channel BN statistics over valid rows of this tile
    float bias = b1[ch];
    float s = 0.f, sq = 0.f;
#pragma unroll
    for (int r = 0; r < 8; r++) {
        int row = rt * 16 + r + 8 * khalf;    // C/D layout: M = r + 8*(lane>=16)
        if (row < ROWS) {
            float v = c[r] + bias;
            s  += v;
            sq += v * v;
        }
    }
    atomicAdd(&st[ch], s);
    atomicAdd(&st[CCH + ch], sq);
    __syncthreads();
    for (int i = tid; i < CCH; i += NTHREADS) {
        atomicAdd(&wsum[i],   st[i]);
        atomicAdd(&wsumsq[i], st[CCH + i]);
    }
}

// Fold BN into scale/shift: h_bn = h*scale + shift
__global__ void k_wfin(const float* __restrict__ wsum, const float* __restrict__ wsumsq,
                       const float* __restrict__ g1, const float* __restrict__ be1,
                       float* __restrict__ wscale, float* __restrict__ wshift)
{
    int c = threadIdx.x;                      // 192 threads
    const float cnt = (float)B_ * (float)ROWS;
    float mu  = wsum[c] / cnt;
    float var = wsumsq[c] / cnt - mu * mu;
    float inv = rsqrtf(var + EPS_);
    float sc  = inv * g1[c];
    wscale[c] = sc;
    wshift[c] = be1[c] - mu * sc;
}

// ---------------------------------------------------------------------------
// Pass 2: one block per batch element. Recompute H tiles with WMMA, apply
// folded BN + leaky-relu, reduce with w2 into s(i,j) in LDS, masked softmax,
// then the whole gconv (agg = attn@x, concat-GEMM) without leaving LDS.
// For the last layer write logits of node 0 straight to d_out.
// ---------------------------------------------------------------------------
__global__ void __launch_bounds__(NTHREADS)
k_attn(const float* __restrict__ X, const float* __restrict__ w1,
       const float* __restrict__ b1, const float* __restrict__ wscale,
       const float* __restrict__ wshift, const float* __restrict__ w2,
       const float* __restrict__ fw, const float* __restrict__ fb,
       int fin, int fout, int isFinal,
       float* __restrict__ hpre, float* __restrict__ gsum,
       float* __restrict__ gsumsq, float* __restrict__ out)
{
    __shared__ float Xs[N_ * (XMAX + 4)];
    __shared__ float srow[ROWS];
    __shared__ float attnS[ROWS];
    __shared__ float P[N_ * HALF_];           // x @ fw_right, per node
    __shared__ float gpart[2 * HALF_];

    const int tid = threadIdx.x;
    const int b   = blockIdx.x;
    const int xstr = fin + 4;

    const float* xb = X + (size_t)b * N_ * XMAX;
    for (int idx = tid; idx < N_ * fin; idx += NTHREADS) {
        int n = idx / fin, f = idx - n * fin;
        Xs[n * xstr + f] = xb[n * XMAX + f];
    }
    for (int i = tid; i < ROWS; i += NTHREADS) srow[i] = 0.f;
    for (int i = tid; i < 2 * HALF_; i += NTHREADS) gpart[i] = 0.f;
    __syncthreads();

    const int wave  = tid >> 5;
    const int lane  = tid & 31;
    const int l16   = lane & 15;
    const int khalf = lane >> 4;
    const int ch    = wave * 16 + l16;
    const float* wrow = w1 + (size_t)ch * fin;
    const float bias = b1[ch], sc = wscale[ch], sh = wshift[ch], w2c = w2[ch];

    for (int rt = 0; rt < RT; rt++) {
        int grow = rt * 16 + l16;
        int gr   = (grow < ROWS) ? grow : 0;
        int ii   = gr / N_, jj = gr - ii * N_;
        const float* xi = Xs + ii * xstr;
        const float* xj = Xs + jj * xstr;

        v8f c = {0.f, 0.f, 0.f, 0.f, 0.f, 0.f, 0.f, 0.f};
        for (int k = 0; k < fin; k += 4) {
            int f0 = k + 2 * khalf;
            v2f a, bb;
            a.x  = fabsf(xi[f0]     - xj[f0]);
            a.y  = fabsf(xi[f0 + 1] - xj[f0 + 1]);
            bb.x = wrow[f0];
            bb.y = wrow[f0 + 1];
            c = __builtin_amdgcn_wmma_f32_16x16x4_f32(false, a, false, bb,
                                                      (short)0, c, false, false);
        }
#pragma unroll
        for (int r = 0; r < 8; r++) {
            int row = rt * 16 + r + 8 * khalf;
            if (row < ROWS) {
                float v = (c[r] + bias) * sc + sh;    // BN folded
                v = (v > 0.f) ? v : SLOPE * v;        // leaky relu
                atomicAdd(&srow[row], v * w2c);       // reduce over channels
            }
        }
    }
    __syncthreads();

    // masked softmax over j (diagonal -> 0; softmax is shift-invariant so b2
    // cancels and is omitted)
    if (tid < N_) {
        int i = tid;
        float mx = -1e30f;
        for (int j = 0; j < N_; j++)
            if (j != i) mx = fmaxf(mx, srow[i * N_ + j]);
        float sum = 0.f;
        for (int j = 0; j < N_; j++) {
            float e = (j == i) ? 0.f : expf(srow[i * N_ + j] - mx);
            attnS[i * N_ + j] = e;
            sum += e;
        }
        float inv = 1.f / sum;
        for (int j = 0; j < N_; j++) attnS[i * N_ + j] *= inv;
    }
    __syncthreads();

    // P[j][o] = sum_f Xs[j][f] * fw[o][fin + f]   (right half of concat GEMM)
    for (int p = tid; p < N_ * fout; p += NTHREADS) {
        int j = p / fout, o = p - j * fout;
        const float* fr = fw + (size_t)o * (2 * fin) + fin;
        const float* xr = Xs + j * xstr;
        float acc = 0.f;
        for (int f = 0; f < fin; f++) acc += xr[f] * fr[f];
        P[j * fout + o] = acc;
    }
    __syncthreads();

    // h[n][o] = x[n]·fw_left[o] + sum_j attn[n][j] * P[j][o] + fb[o]
    for (int p = tid; p < N_ * fout; p += NTHREADS) {
        int n = p / fout, o = p - n * fout;
        const float* fr = fw + (size_t)o * (2 * fin);
        const float* xr = Xs + n * xstr;
        float acc = fb[o];
        for (int f = 0; f < fin; f++) acc += xr[f] * fr[f];
        for (int j = 0; j < N_; j++) acc += attnS[n * N_ + j] * P[j * fout + o];
        if (isFinal) {
            if (n == 0) out[b * fout + o] = acc;     // logits of query node
        } else {
            hpre[((size_t)b * N_ + n) * HALF_ + o] = acc;
            atomicAdd(&gpart[o], acc);
            atomicAdd(&gpart[HALF_ + o], acc * acc);
        }
    }
    __syncthreads();
    if (!isFinal) {
        for (int i = tid; i < HALF_; i += NTHREADS) {
            atomicAdd(&gsum[i],   gpart[i]);
            atomicAdd(&gsumsq[i], gpart[HALF_ + i]);
        }
    }
}

__global__ void k_gfin(const float* __restrict__ gsum, const float* __restrict__ gsumsq,
                       const float* __restrict__ g, const float* __restrict__ be,
                       float* __restrict__ gscale, float* __restrict__ gshift)
{
    int c = threadIdx.x;                      // 48 threads
    const float cnt = (float)B_ * (float)N_;
    float mu  = gsum[c] / cnt;
    float var = gsumsq[c] / cnt - mu * mu;
    float inv = rsqrtf(var + EPS_);
    float sc  = inv * g[c];
    gscale[c] = sc;
    gshift[c] = be[c] - mu * sc;
}

__global__ void k_gapply(const float* __restrict__ hpre,
                         const float* __restrict__ gscale,
                         const float* __restrict__ gshift,
                         float* __restrict__ X, int fin)
{
    int idx = blockIdx.x * blockDim.x + threadIdx.x;
    if (idx < B_ * N_ * HALF_) {
        int o  = idx % HALF_;
        int bn = idx / HALF_;
        float v = hpre[idx] * gscale[o] + gshift[o];
        v = (v > 0.f) ? v : SLOPE * v;
        X[(size_t)bn * XMAX + fin + o] = v;   // append features in place
    }
}

// ---------------------------------------------------------------------------
extern "C" void kernel_launch(void* const* d_in, const int* in_sizes, int n_in,
                              void* d_out, int out_size, void* d_ws, size_t ws_size,
                              hipStream_t stream)
{
    (void)in_sizes; (void)n_in; (void)out_size; (void)ws_size;

    // setup_inputs() insertion order:
    //   x, {w[0..3]: w1,b1,g1,be1,w2,b2}, {l[0..3]: fw,fb,g,be},
    //   wl: w1,b1,g1,be1,w2,b2, ll: fw,fb
    const float* x = (const float*)d_in[0];
    const float *W1[5], *B1[5], *G1[5], *BE1[5], *W2[5];
    for (int i = 0; i < 4; i++) {
        int b0 = 1 + 6 * i;
        W1[i]  = (const float*)d_in[b0 + 0];
        B1[i]  = (const float*)d_in[b0 + 1];
        G1[i]  = (const float*)d_in[b0 + 2];
        BE1[i] = (const float*)d_in[b0 + 3];
        W2[i]  = (const float*)d_in[b0 + 4];
    }
    W1[4]  = (const float*)d_in[41];
    B1[4]  = (const float*)d_in[42];
    G1[4]  = (const float*)d_in[43];
    BE1[4] = (const float*)d_in[44];
    W2[4]  = (const float*)d_in[45];

    const float *FW[5], *FB[5], *GG[4], *GBE[4];
    for (int i = 0; i < 4; i++) {
        int b0 = 25 + 4 * i;
        FW[i]  = (const float*)d_in[b0 + 0];
        FB[i]  = (const float*)d_in[b0 + 1];
        GG[i]  = (const float*)d_in[b0 + 2];
        GBE[i] = (const float*)d_in[b0 + 3];
    }
    FW[4] = (const float*)d_in[47];
    FB[4] = (const float*)d_in[48];

    // workspace layout (floats)
    float* X      = (float*)d_ws;                       // 512*26*320
    float* hpre   = X + (size_t)B_ * N_ * XMAX;         // 512*26*48
    float* stats  = hpre + (size_t)B_ * N_ * HALF_;     // 960 floats
    float* wsum   = stats;
    float* wsumsq = stats + 192;
    float* wscale = stats + 384;
    float* wshift = stats + 576;
    float* gsum   = stats + 768;
    float* gsumsq = stats + 816;
    float* gscale = stats + 864;
    float* gshift = stats + 912;

    k_copy<<<(B_ * N_ * F0_ + 255) / 256, 256, 0, stream>>>(x, X);

    for (int i = 0; i < 5; i++) {
        const int fin = F0_ + HALF_ * i;                // 128..320
        const int isFinal = (i == 4);
        const int fout = isFinal ? 5 : HALF_;

        k_zero<<<4, 256, 0, stream>>>(stats, 960);
        k_wstats<<<B_ * RT, NTHREADS, 0, stream>>>(X, W1[i], B1[i], fin,
                                                   wsum, wsumsq);
        k_wfin<<<1, 192, 0, stream>>>(wsum, wsumsq, G1[i], BE1[i],
                                      wscale, wshift);
        k_attn<<<B_, NTHREADS, 0, stream>>>(X, W1[i], B1[i], wscale, wshift,
                                            W2[i], FW[i], FB[i], fin, fout,
                                            isFinal, hpre, gsum, gsumsq,
                                            (float*)d_out);
        if (!isFinal) {
            k_gfin<<<1, 48, 0, stream>>>(gsum, gsumsq, GG[i], GBE[i],
                                         gscale, gshift);
            k_gapply<<<(B_ * N_ * HALF_ + 255) / 256, 256, 0, stream>>>(
                hpre, gscale, gshift, X, fin);
        }
    }
}